// UnifiedCADLoss_79611513799125
// MI455X (gfx1250) — compile-verified
//
#include <hip/hip_runtime.h>
#include <cstdint>
#include <cstddef>

// ---------------------------------------------------------------------------
// UnifiedCADLoss on gfx1250 (MI455X)
//
// Roofline: dominant cost = streaming 1 GiB of args logits once (~46us at
// 23.3 TB/s). No matmul structure -> WMMA not applicable; the CDNA5 path that
// matters is async global->LDS (ASYNCcnt) double-buffered streaming so the
// per-element v_exp_f32 of the online logsumexp overlaps HBM traffic, plus
// early row-skipping on the combined mask (most rows are masked out).
// ---------------------------------------------------------------------------

#define EOS_IDX 3
#define TOL 3
#define ALPHA 2.0f

constexpr int kB = 4, kS = 512, kC = 10, kN = 16, kV = 8192;
constexpr int kRows  = kB * kS * kN;     // 32768 rows of length V
constexpr int kChunk = 512;              // floats per streamed chunk (2 KB)
constexpr int kNChunk = kV / kChunk;     // 16
constexpr int kWavesPerBlock = 8;        // 256 threads (wave32)

// ---------------------------------------------------------------------------
// Issue one chunk (512 floats) as 4 async b128 loads into LDS.
// GVS addressing: mem = SGPR base + signed 32-bit VGPR byte offset.
// Max byte offset = 32768*8192*4 = 2^30 < 2^31, fits I32.
// ---------------------------------------------------------------------------
__device__ __forceinline__ void issue_chunk_async(const float* __restrict__ base,
                                                  long byte_off, unsigned lds_base,
                                                  int lane) {
#pragma unroll
  for (int j = 0; j < 4; ++j) {
    int voff = (int)(byte_off + (long)(j * 128 + lane * 4) * 4);
    unsigned laddr = lds_base + (unsigned)((j * 32 + lane) * 16);
    asm volatile("global_load_async_to_lds_b128 %0, %1, %2"
                 :: "v"(laddr), "v"(voff), "s"(base)
                 : "memory");
  }
}

// ---------------------------------------------------------------------------
// Kernel 1: valid mask (positions up to & including first EOS) + zero accum.
// One wave32 per batch row; ballot prefix-scan over S in chunks of 32.
// ws layout: ws[0..3] = {cmd_num, cmd_den, args_num, args_den}; ws[4..] mask.
// ---------------------------------------------------------------------------
__global__ void vmask_init_kernel(const int* __restrict__ commands,
                                  float* __restrict__ ws) {
  float* acc   = ws;
  float* vmask = ws + 4;
  const int b = blockIdx.x;
  const int lane = threadIdx.x;
  if (b == 0 && lane < 4) acc[lane] = 0.0f;

  int carry = 0;
  for (int s0 = 0; s0 < kS; s0 += 32) {
    const int s = s0 + lane;
    const int cmd = commands[b * kS + s];
    const unsigned bal = (unsigned)__ballot(cmd == EOS_IDX);   // wave32: low bits
    const unsigned incl = bal & (lane == 31 ? 0xFFFFFFFFu : ((1u << (lane + 1)) - 1u));
    const int cum = carry + __popc(incl);                       // inclusive cumsum
    vmask[b * kS + s] = (cum <= 1) ? 1.0f : 0.0f;
    carry += __popc(bal);
  }
}

// ---------------------------------------------------------------------------
// Kernel 2: command cross-entropy (tiny: 2048 positions x 10 classes).
// ---------------------------------------------------------------------------
__global__ void cmd_loss_kernel(const float* __restrict__ logits,
                                const int* __restrict__ commands,
                                const float* __restrict__ vmask,
                                float* __restrict__ acc) {
  const int pos = blockIdx.x * blockDim.x + threadIdx.x;
  float nll = 0.0f, valid = 0.0f;
  if (pos < kB * kS) {
    const float* row = logits + pos * kC;
    float mx = row[0];
#pragma unroll
    for (int i = 1; i < kC; ++i) mx = fmaxf(mx, row[i]);
    float se = 0.0f;
#pragma unroll
    for (int i = 0; i < kC; ++i) se += __expf(row[i] - mx);
    int c = commands[pos];
    c = min(max(c, 0), kC - 1);
    nll = -(row[c] - mx - __logf(se));
    if (nll != nll) nll = 0.0f;                  // NaN guard per reference
    valid = vmask[pos];
    nll *= valid;
  }
#pragma unroll
  for (int off = 16; off; off >>= 1) {
    nll   += __shfl_xor(nll, off, 32);
    valid += __shfl_xor(valid, off, 32);
  }
  if ((threadIdx.x & 31) == 0) {
    atomicAdd(&acc[0], nll);
    atomicAdd(&acc[1], valid);
  }
}

// ---------------------------------------------------------------------------
// Kernel 3: args loss. One wave per [b,s,n] row of V=8192.
//  - skip row if combined mask == 0 (no HBM traffic)
//  - stream 16x512-float chunks through LDS via async-to-LDS, double buffered
//  - per-lane online (max, sumexp), butterfly merge, 7-point weighted gather
// ---------------------------------------------------------------------------
__global__ __launch_bounds__(256) void args_loss_kernel(
    const float* __restrict__ logits,     // [Rows, V]
    const float* __restrict__ args_mask,  // [C, N]
    const int*  __restrict__ commands,    // [B*S]
    const int*  __restrict__ tokens,      // [Rows]
    const float* __restrict__ vmask,      // [B*S]
    float* __restrict__ acc) {
  __shared__ float4 stage[kWavesPerBlock][2][kChunk / 4];   // 8*2*128*16B = 32 KB

  const int lane  = threadIdx.x & 31;
  const int w     = threadIdx.x >> 5;
  const int gwave = (blockIdx.x * blockDim.x + threadIdx.x) >> 5;
  const int nwave = (gridDim.x * blockDim.x) >> 5;

  // Soft-label weights (identical on all lanes).
  float wsum = 0.0f;
#pragma unroll
  for (int k = 0; k < 2 * TOL + 1; ++k)
    wsum += __expf(-ALPHA * fabsf((float)(k - TOL)));
  const float inv  = 1.0f / (wsum + 1e-8f);
  const float myw  = (lane < 2 * TOL + 1)
                       ? __expf(-ALPHA * fabsf((float)(lane - TOL))) * inv : 0.0f;
  const float wtot = wsum * inv;                     // sum of normalized weights

  const unsigned lds0 = (unsigned)(size_t)&stage[w][0][0];  // LDS byte address
  const unsigned lds1 = (unsigned)(size_t)&stage[w][1][0];

  float lnum = 0.0f, lden = 0.0f;

  for (int r = gwave; r < kRows; r += nwave) {
    const int pos = r / kN;                 // b*S + s
    const int n   = r - pos * kN;
    int cmd = commands[pos];
    cmd = min(max(cmd, 0), kC - 1);
    const float m = vmask[pos] * args_mask[cmd * kN + n];   // uniform over wave
    if (m <= 0.0f) continue;                // dead row: skip all HBM traffic

    const float* rowp = logits + (size_t)r * kV;
    const long rowbyte = (long)r * kV * 4;

    // Prime the pipeline: chunk 0 -> buffer 0.
    issue_chunk_async(logits, rowbyte, lds0, lane);

    float mx = -INFINITY, sm = 0.0f;        // per-lane online logsumexp state
    for (int c = 0; c < kNChunk; ++c) {
      if (c + 1 < kNChunk) {
        issue_chunk_async(logits, rowbyte + (long)(c + 1) * kChunk * 4,
                          ((c + 1) & 1) ? lds1 : lds0, lane);
        asm volatile("s_wait_asynccnt 4" ::: "memory");  // chunk c complete
      } else {
        asm volatile("s_wait_asynccnt 0" ::: "memory");
      }
      const float4* buf = stage[w][c & 1];
      float4 x[4];
      float vmaxl = -INFINITY;
#pragma unroll
      for (int k = 0; k < 4; ++k) {
        x[k] = buf[k * 32 + lane];          // ds_load_b128, conflict-free
        vmaxl = fmaxf(vmaxl, fmaxf(fmaxf(x[k].x, x[k].y), fmaxf(x[k].z, x[k].w)));
      }
      const float M = fmaxf(mx, vmaxl);
      float a = 0.0f;
#pragma unroll
      for (int k = 0; k < 4; ++k) {
        a += __expf(x[k].x - M) + __expf(x[k].y - M)
           + __expf(x[k].z - M) + __expf(x[k].w - M);
      }
      sm = sm * __expf(mx - M) + a;         // rescale old accumulator
      mx = M;
    }

    // Cross-lane butterfly merge of (mx, sm): every lane ends with row totals.
#pragma unroll
    for (int off = 16; off; off >>= 1) {
      const float mo = __shfl_xor(mx, off, 32);
      const float so = __shfl_xor(sm, off, 32);
      const float M2 = fmaxf(mx, mo);
      sm = sm * __expf(mx - M2) + so * __expf(mo - M2);
      mx = M2;
    }
    const float logZ = mx + __logf(sm);

    // 7-point weighted gather around target (lanes 0..6), butterfly sum.
    int tgt = tokens[r];
    tgt = min(max(tgt, 0), kV - 1);
    float g = 0.0f;
    if (lane < 2 * TOL + 1) {
      const int idx = min(max(tgt + lane - TOL, 0), kV - 1);
      g = myw * rowp[idx];                  // hits L2 (row just streamed)
    }
#pragma unroll
    for (int off = 16; off; off >>= 1) g += __shfl_xor(g, off, 32);

    float loss = -g + logZ * wtot;          // = -(sum w_k * logp[idx_k])
    if (loss != loss) loss = 0.0f;          // NaN guard per reference
    lnum += loss * m;
    lden += m;
  }

  if (lane == 0) {                          // lnum/lden uniform across lanes
    atomicAdd(&acc[2], lnum);
    atomicAdd(&acc[3], lden);
  }
}

// ---------------------------------------------------------------------------
// Kernel 4: finalize -> (total, loss_cmd, loss_args)
// ---------------------------------------------------------------------------
__global__ void finalize_kernel(const float* __restrict__ acc,
                                float* __restrict__ out) {
  if (threadIdx.x == 0 && blockIdx.x == 0) {
    const float lc = acc[0] / (acc[1] + 1e-8f);
    const float la = acc[2] / (acc[3] + 1e-8f);
    out[0] = 1.0f * lc + 1.0f * la;   // W_CMD*loss_cmd + W_ARGS*loss_args
    out[1] = lc;
    out[2] = la;
  }
}

// ---------------------------------------------------------------------------
extern "C" void kernel_launch(void* const* d_in, const int* in_sizes, int n_in,
                              void* d_out, int out_size, void* d_ws, size_t ws_size,
                              hipStream_t stream) {
  const float* cmd_logits  = (const float*)d_in[0];  // [B,S,C]
  const float* args_logits = (const float*)d_in[1];  // [B,S,N,V]
  const float* args_mask   = (const float*)d_in[2];  // [C,N]
  const int*   commands    = (const int*)d_in[3];    // [B,S]
  const int*   tokens      = (const int*)d_in[4];    // [B,S,N]
  float* ws  = (float*)d_ws;   // [0..3] accum, [4..4+B*S) valid mask
  float* out = (float*)d_out;  // 3 floats

  vmask_init_kernel<<<kB, 32, 0, stream>>>(commands, ws);
  cmd_loss_kernel<<<(kB * kS + 255) / 256, 256, 0, stream>>>(
      cmd_logits, commands, ws + 4, ws);
  args_loss_kernel<<<512, 256, 0, stream>>>(
      args_logits, args_mask, commands, tokens, ws + 4, ws);
  finalize_kernel<<<1, 32, 0, stream>>>(ws, out);
}